// MultiHeadAttention_50130858279186
// MI455X (gfx1250) — compile-verified
//
#include <hip/hip_runtime.h>

typedef float v2f __attribute__((ext_vector_type(2)));
typedef float v8f __attribute__((ext_vector_type(8)));

#define D_MODEL 128

// ---------------------------------------------------------------------------
// Kernel A: fused Q/K/V projection.  One wave32 per 16x16 output tile,
// V_WMMA_F32_16X16X4_F32 accumulating over K=128 in chunks of 4.
// grid = (N/16, 3), block = 256 (8 waves -> 8 column tiles of 16).
// ---------------------------------------------------------------------------
__global__ __launch_bounds__(256)
void qkv_proj_kernel(const float* __restrict__ x,
                     const float* __restrict__ Wq, const float* __restrict__ bq,
                     const float* __restrict__ Wk, const float* __restrict__ bk,
                     const float* __restrict__ Wv, const float* __restrict__ bv,
                     float* __restrict__ qo, float* __restrict__ ko,
                     float* __restrict__ vo)
{
    const int lane = threadIdx.x & 31;
    const int wave = threadIdx.x >> 5;      // column tile 0..7
    const int rowTile = blockIdx.x;

    const float* W; const float* bias; float* out;
    if (blockIdx.y == 0)      { W = Wq; bias = bq; out = qo; }
    else if (blockIdx.y == 1) { W = Wk; bias = bk; out = ko; }
    else                      { W = Wv; bias = bv; out = vo; }

    const int m    = lane & 15;             // A row within tile / C column lane
    const int half = lane >> 4;             // K sub-slot (A) / row half (C)
    const int n    = wave * 16 + m;         // output column

    v8f acc;
    const float bn = bias[n];
#pragma unroll
    for (int r = 0; r < 8; ++r) acc[r] = bn;

    const float* xrow = x + (size_t)(rowTile * 16 + m) * D_MODEL;

#pragma unroll 4
    for (int kk = 0; kk < 32; ++kk) {
        const int k0 = kk * 4 + half * 2;
        v2f a;  a.x  = xrow[k0];
                a.y  = xrow[k0 + 1];
        v2f bf; bf.x = W[(size_t)k0 * D_MODEL + n];
                bf.y = W[(size_t)(k0 + 1) * D_MODEL + n];
        acc = __builtin_amdgcn_wmma_f32_16x16x4_f32(
                  false, a, false, bf, (short)0, acc, false, false);
    }

    // C/D layout: lane<16 -> rows 0..7, lane>=16 -> rows 8..15 (VGPR r = row half*8+r)
    float* op = out + (size_t)(rowTile * 16 + half * 8) * D_MODEL + n;
#pragma unroll
    for (int r = 0; r < 8; ++r) op[(size_t)r * D_MODEL] = acc[r];
}

// ---------------------------------------------------------------------------
// Kernel B: edge phase.  One wave32 per edge; lane handles 4 contiguous
// channels (float4 gathers of L2-resident q/k/v rows), native f32 atomics
// into L2-resident z/num accumulators.
// ---------------------------------------------------------------------------
__global__ __launch_bounds__(256)
void edge_attn_kernel(const float* __restrict__ q, const float* __restrict__ k,
                      const float* __restrict__ v,
                      const int* __restrict__ src, const int* __restrict__ dst,
                      float* __restrict__ z, float* __restrict__ num, int E)
{
    const int e = blockIdx.x * 8 + (threadIdx.x >> 5);
    if (e >= E) return;
    const int lane = threadIdx.x & 31;
    const int s = src[e];
    const int d = dst[e];
    const float inv_sqrt_dk = 0.25f;        // 1/sqrt(16)

    const float4 kv = *(const float4*)(k + (size_t)s * D_MODEL + lane * 4);
    const float4 qv = *(const float4*)(q + (size_t)d * D_MODEL + lane * 4);
    const float4 vv = *(const float4*)(v + (size_t)s * D_MODEL + lane * 4);

    const float m0 = __expf(kv.x * qv.x * inv_sqrt_dk);
    const float m1 = __expf(kv.y * qv.y * inv_sqrt_dk);
    const float m2 = __expf(kv.z * qv.z * inv_sqrt_dk);
    const float m3 = __expf(kv.w * qv.w * inv_sqrt_dk);

    float* zp = z   + (size_t)d * D_MODEL + lane * 4;
    float* np = num + (size_t)d * D_MODEL + lane * 4;
    unsafeAtomicAdd(zp + 0, m0);
    unsafeAtomicAdd(zp + 1, m1);
    unsafeAtomicAdd(zp + 2, m2);
    unsafeAtomicAdd(zp + 3, m3);
    unsafeAtomicAdd(np + 0, m0 * vv.x);
    unsafeAtomicAdd(np + 1, m1 * vv.y);
    unsafeAtomicAdd(np + 2, m2 * vv.z);
    unsafeAtomicAdd(np + 3, m3 * vv.w);
}

// ---------------------------------------------------------------------------
// Kernel C: fused normalize (num/z) + output projection @ Wo + bo.
// Same WMMA tiling as kernel A; NaN (0/0, isolated nodes) propagates through
// WMMA exactly like the reference.
// ---------------------------------------------------------------------------
__global__ __launch_bounds__(256)
void out_proj_kernel(const float* __restrict__ num, const float* __restrict__ z,
                     const float* __restrict__ Wo, const float* __restrict__ bo,
                     float* __restrict__ out)
{
    const int lane = threadIdx.x & 31;
    const int wave = threadIdx.x >> 5;
    const int rowTile = blockIdx.x;

    const int m    = lane & 15;
    const int half = lane >> 4;
    const int n    = wave * 16 + m;

    v8f acc;
    const float bn = bo[n];
#pragma unroll
    for (int r = 0; r < 8; ++r) acc[r] = bn;

    const float* nr = num + (size_t)(rowTile * 16 + m) * D_MODEL;
    const float* zr = z   + (size_t)(rowTile * 16 + m) * D_MODEL;

#pragma unroll 4
    for (int kk = 0; kk < 32; ++kk) {
        const int k0 = kk * 4 + half * 2;
        v2f a;  a.x  = nr[k0]     / zr[k0];
                a.y  = nr[k0 + 1] / zr[k0 + 1];
        v2f bf; bf.x = Wo[(size_t)k0 * D_MODEL + n];
                bf.y = Wo[(size_t)(k0 + 1) * D_MODEL + n];
        acc = __builtin_amdgcn_wmma_f32_16x16x4_f32(
                  false, a, false, bf, (short)0, acc, false, false);
    }

    float* op = out + (size_t)(rowTile * 16 + half * 8) * D_MODEL + n;
#pragma unroll
    for (int r = 0; r < 8; ++r) op[(size_t)r * D_MODEL] = acc[r];
}

// ---------------------------------------------------------------------------
extern "C" void kernel_launch(void* const* d_in, const int* in_sizes, int n_in,
                              void* d_out, int out_size, void* d_ws, size_t ws_size,
                              hipStream_t stream)
{
    const float* x   = (const float*)d_in[0];
    const int*   src = (const int*)  d_in[1];
    const int*   dst = (const int*)  d_in[2];
    const float* Wq  = (const float*)d_in[3];
    const float* bq  = (const float*)d_in[4];
    const float* Wk  = (const float*)d_in[5];
    const float* bk  = (const float*)d_in[6];
    const float* Wv  = (const float*)d_in[7];
    const float* bv  = (const float*)d_in[8];
    const float* Wo  = (const float*)d_in[9];
    const float* bo  = (const float*)d_in[10];

    const int N = in_sizes[0] / D_MODEL;   // 50000 (multiple of 16)
    const int E = in_sizes[1];             // 800000

    // Workspace: q, k, v, z, num — 5 * N * 128 floats = 128 MB (L2-resident)
    float* q  = (float*)d_ws;
    float* k  = q  + (size_t)N * D_MODEL;
    float* v  = k  + (size_t)N * D_MODEL;
    float* z  = v  + (size_t)N * D_MODEL;
    float* nm = z  + (size_t)N * D_MODEL;

    // Zero the segment-sum accumulators (graph-capture legal).
    hipMemsetAsync(z, 0, (size_t)2 * N * D_MODEL * sizeof(float), stream);

    dim3 gridA(N / 16, 3);
    qkv_proj_kernel<<<gridA, 256, 0, stream>>>(x, Wq, bq, Wk, bk, Wv, bv, q, k, v);

    edge_attn_kernel<<<(E + 7) / 8, 256, 0, stream>>>(q, k, v, src, dst, z, nm, E);

    out_proj_kernel<<<N / 16, 256, 0, stream>>>(nm, z, Wo, bo, (float*)d_out);
}